// ModelNew_17411797418168
// MI455X (gfx1250) — compile-verified
//
#include <hip/hip_runtime.h>

// Speculative-decode step preparation (pure int gather/scatter, memory-latency bound).
// One thread per output token; all four output streams written coalesced; branchless.
//
// out layout (floats, concatenated in reference return order):
//   [0*n, 1*n): out_tokens     [1*n, 2*n): pos
//   [2*n, 3*n): out_seq_lens   [3*n, 4*n): slots
//
// POW2T: T is a power of two (host-known), so r/j come from shift/mask.
// block_size is a device scalar (unknown to host under graph capture); the
// power-of-two fast path is selected by a wave-uniform branch in-kernel.

template <bool POW2T>
__global__ __launch_bounds__(256) void specdec_prepare_kernel(
    const int* __restrict__ sampled_tokens,   // [R, SC]
    const int* __restrict__ input_positions,  // [R*T]
    const int* __restrict__ block_table,      // [R, MAX_BLOCKS]
    const int* __restrict__ spec_tokens,      // [R, SPEC]
    const int* __restrict__ accepted_num,     // [R]
    const int* __restrict__ block_size_p,     // scalar (device)
    unsigned T, unsigned Tshift,              // Tshift valid iff POW2T
    int SC, int SPEC, int MAX_BLOCKS,
    float* __restrict__ out, unsigned n)      // n = R*T
{
    const unsigned i = blockIdx.x * blockDim.x + threadIdx.x;
    if (i >= n) return;

    unsigned r, j;
    if (POW2T) {
        r = i >> Tshift;
        j = i & (T - 1u);
    } else {
        r = i / T;            // unsigned: no sign-fixup sequence
        j = i - r * T;
    }

    const unsigned bs = (unsigned)block_size_p[0];

    // a = clip(accepted_num[r], 1, SC)
    int a = accepted_num[r];
    a = a < 1 ? 1 : (a > SC ? SC : a);

    // base position of the new first token; r*T == i - j
    const int prev_base = input_positions[i - j];
    const unsigned pos = (unsigned)(prev_base + a) + j;

    // Branchless token select: always gather both candidates with safe indices,
    // then v_cndmask. Avoids a divergent saveexec branch (4 lanes/wave take j==0).
    const int stok = sampled_tokens[r * SC + (unsigned)(a - 1)];
    const int dtok = spec_tokens[r * SPEC + (j == 0u ? 0u : j - 1u)];
    const int tok  = (j == 0u) ? stok : dtok;

    // slot via page table gather; bs is wave-uniform -> uniform branch
    unsigned bidx, boff;
    if ((bs & (bs - 1u)) == 0u) {                 // power-of-two block size
        const unsigned sh = 31u - (unsigned)__clz((int)bs);
        bidx = pos >> sh;
        boff = pos & (bs - 1u);
    } else {
        bidx = pos / bs;
        boff = pos - bidx * bs;
    }
    const int blk  = block_table[(size_t)r * (size_t)MAX_BLOCKS + (size_t)bidx];
    const unsigned slot = (unsigned)blk * bs + boff;

    out[i]          = (float)tok;           // out_tokens
    out[n + i]      = (float)pos;           // positions
    out[2u * n + i] = (float)(pos + 1u);    // out_seq_lens
    out[3u * n + i] = (float)slot;          // slots
}

extern "C" void kernel_launch(void* const* d_in, const int* in_sizes, int n_in,
                              void* d_out, int out_size, void* d_ws, size_t ws_size,
                              hipStream_t stream) {
    // setup_inputs() order:
    // 0 input_tokens (unused)  1 sampled_tokens  2 input_positions  3 seq_lens (unused)
    // 4 slot_mapping (unused)  5 block_table     6 spec_tokens      7 accepted_num
    // 8 num_seqs (scalar)      9 num_queries (scalar)               10 block_size (scalar)
    const int* sampled   = (const int*)d_in[1];
    const int* in_pos    = (const int*)d_in[2];
    const int* btab      = (const int*)d_in[5];
    const int* spec      = (const int*)d_in[6];
    const int* accepted  = (const int*)d_in[7];
    const int* bs_dev    = (const int*)d_in[10];

    const int R          = in_sizes[7];          // accepted_num has R elements
    const unsigned T     = (unsigned)(in_sizes[2] / R);   // input_positions: R*T
    const int SC         = in_sizes[1] / R;      // sampled_tokens: R*SC
    const int SPEC       = in_sizes[6] / R;      // spec_tokens:    R*SPEC
    const int MAX_BLOCKS = in_sizes[5] / R;      // block_table:    R*MAX_BLOCKS
    const unsigned n     = (unsigned)R * T;      // tokens emitted (out_size == 4*n)

    float* out = (float*)d_out;

    const int threads = 256;
    const int blocks  = (int)((n + threads - 1) / threads);

    const bool pow2T = (T & (T - 1u)) == 0u;
    unsigned tshift = 0;
    { unsigned t = T; while (t > 1u) { t >>= 1u; ++tshift; } }

    if (pow2T) {
        specdec_prepare_kernel<true><<<blocks, threads, 0, stream>>>(
            sampled, in_pos, btab, spec, accepted, bs_dev,
            T, tshift, SC, SPEC, MAX_BLOCKS, out, n);
    } else {
        specdec_prepare_kernel<false><<<blocks, threads, 0, stream>>>(
            sampled, in_pos, btab, spec, accepted, bs_dev,
            T, tshift, SC, SPEC, MAX_BLOCKS, out, n);
    }
}